// YoloLoss_71004399337999
// MI455X (gfx1250) — compile-verified
//
#include <hip/hip_runtime.h>
#include <math.h>

#define NO    86
#define NCLS  80
#define BSZ   16
#define NTGT  32
#define KNEG  1000
#define NNEG  (3 * BSZ * KNEG)        // 48000
#define POS_BLOCKS_PER_LVL 64          // 512 targets / 8 waves per block
#define NEG_BLOCKS ((NNEG + 255) / 256) // 188
#define NPART (3 * POS_BLOCKS_PER_LVL + NEG_BLOCKS) // 380

__constant__ float c_anch[3][3][2] = {
    {{10.f, 13.f}, {16.f, 30.f}, {33.f, 23.f}},
    {{30.f, 61.f}, {62.f, 45.f}, {59.f, 119.f}},
    {{116.f, 90.f}, {156.f, 198.f}, {373.f, 326.f}}};

__device__ __forceinline__ float softplusf(float x) {
    // jax.nn.softplus(x) == logaddexp(x, 0) == max(x,0) + log1p(exp(-|x|))
    return fmaxf(x, 0.0f) + log1pf(expf(-fabsf(x)));
}

__device__ __forceinline__ float smooth_l1(float d) {
    float a = fabsf(d);
    return (a < 1.0f) ? 0.5f * a * a : a - 0.5f;
}

// ---------------------------------------------------------------------------
// Positive targets: one wave32 per (b, t). Lanes split the 80-class softplus
// loop; lane 0 handles the scalar box/obj/one-hot terms. Weighted partials
// are block-reduced in LDS (fixed order -> deterministic).
// ---------------------------------------------------------------------------
__global__ __launch_bounds__(256) void yolo_pos(
    const float* __restrict__ p, const float* __restrict__ boxes,
    const int* __restrict__ labels, const int* __restrict__ strides,
    int level, int h, int w, float* __restrict__ partials) {
    const int tid  = threadIdx.x;
    const int lane = tid & 31;
    const int wave = tid >> 5;
    const int tgt  = blockIdx.x * 8 + wave;  // 0..511
    const int b    = tgt >> 5;
    const int t    = tgt & 31;

    const float s     = (float)strides[level];
    const float inv_s = 1.0f / s;

    const float* bb = boxes + (size_t)(b * NTGT + t) * 5;
    const float bx = bb[0] * inv_s;
    const float by = bb[1] * inv_s;
    const float bw = bb[2] * inv_s;
    const float bh = bb[3] * inv_s;
    const float ta = bb[4];

    // argmin over 3 anchors of max aspect ratio (first index wins ties)
    int best = 0;
    float bar = 3.402823466e+38f;
    float abx = 1.0f, aby = 1.0f;
    for (int a = 0; a < 3; ++a) {
        float ax = c_anch[level][a][0] * inv_s;
        float ay = c_anch[level][a][1] * inv_s;
        float rx = bw / ax, ry = bh / ay;
        float ar = fmaxf(fmaxf(rx, 1.0f / rx), fmaxf(ry, 1.0f / ry));
        if (ar < bar) { bar = ar; best = a; abx = ax; aby = ay; }
    }

    int gi = (int)bx;  // truncation toward zero (bx >= 0), matches astype(int32)
    gi = gi < 0 ? 0 : (gi > w - 1 ? w - 1 : gi);
    int gj = (int)by;
    gj = gj < 0 ? 0 : (gj > h - 1 ? h - 1 : gj);

    const size_t base = ((((size_t)b * 3 + best) * h + gj) * w + gi) * NO;

    // class softplus sum, lanes strided across the 80 classes (sel[6..85])
    float cls_sp = 0.0f;
    for (int c = 6 + lane; c < NO; c += 32) cls_sp += softplusf(p[base + c]);
    for (int off = 16; off > 0; off >>= 1) cls_sp += __shfl_down(cls_sp, off, 32);

    float wpart = 0.0f;
    if (lane == 0) {
        const float px   = p[base + 0];
        const float py   = p[base + 1];
        const float pw   = p[base + 2];
        const float ph   = p[base + 3];
        const float pa   = p[base + 4];
        const float pobj = p[base + 5];
        const int lab    = labels[b * NTGT + t];
        const float plab = p[base + 6 + lab];

        const float tx = bx - (float)gi;
        const float ty = by - (float)gj;
        const float tw = fminf(fmaxf(logf(bw / abx + 1e-6f), -4.0f), 4.0f);
        const float th = fminf(fmaxf(logf(bh / aby + 1e-6f), -4.0f), 4.0f);

        const float PI  = 3.14159265358979323846f;
        const float TPI = 6.28318530717958647692f;
        float d = pa - ta + PI;
        float m = fmodf(d, TPI);          // jnp.remainder: sign of divisor
        if (m < 0.0f) m += TPI;
        const float ang = m - PI;

        const float box_l =
            smooth_l1(px - tx) + smooth_l1(py - ty) +
            smooth_l1(fminf(fmaxf(pw, -4.0f), 4.0f) - tw) +
            smooth_l1(fminf(fmaxf(ph, -4.0f), 4.0f) - th) + smooth_l1(ang);
        const float obj_l = softplusf(pobj) - pobj;   // BCE with target 1
        const float cls_l = cls_sp - plab;            // sum softplus - x[label]
        wpart = 0.05f * box_l + 1.0f * obj_l + 0.5f * cls_l;
    }

    __shared__ float wred[8];
    if (lane == 0) wred[wave] = wpart;
    __syncthreads();
    if (tid == 0) {
        float acc = 0.0f;
        for (int i = 0; i < 8; ++i) acc += wred[i];
        partials[blockIdx.x] = acc;
    }
}

// ---------------------------------------------------------------------------
// Negative samples: one thread per (level, b, k); scattered pobj gather +
// softplus, LDS tree block reduce (obj weight = 1.0).
// ---------------------------------------------------------------------------
__global__ __launch_bounds__(256) void yolo_neg(
    const float* __restrict__ p0, const float* __restrict__ p1,
    const float* __restrict__ p2, const int* __restrict__ ri0,
    const int* __restrict__ rj0, const int* __restrict__ rk0,
    const int* __restrict__ ri1, const int* __restrict__ rj1,
    const int* __restrict__ rk1, const int* __restrict__ ri2,
    const int* __restrict__ rj2, const int* __restrict__ rk2,
    float* __restrict__ partials) {
    const int tid = threadIdx.x;
    const int n   = blockIdx.x * 256 + tid;

    float val = 0.0f;
    if (n < NNEG) {
        const int l = n / (BSZ * KNEG);
        const int m = n - l * (BSZ * KNEG);
        const int b = m / KNEG;
        const int k = m - b * KNEG;

        const float* p;
        const int *ri, *rj, *rk;
        int hw;
        if (l == 0)      { p = p0; ri = ri0; rj = rj0; rk = rk0; hw = 80; }
        else if (l == 1) { p = p1; ri = ri1; rj = rj1; rk = rk1; hw = 40; }
        else             { p = p2; ri = ri2; rj = rj2; rk = rk2; hw = 20; }

        const int fi = b * KNEG + k;
        const int a = ri[fi], j = rj[fi], x = rk[fi];
        const size_t idx = ((((size_t)b * 3 + a) * hw + j) * hw + x) * NO + 5;
        val = softplusf(p[idx]);  // BCE with target 0
    }

    __shared__ float red[256];
    red[tid] = val;
    __syncthreads();
    for (int s2 = 128; s2 > 0; s2 >>= 1) {
        if (tid < s2) red[tid] += red[tid + s2];
        __syncthreads();
    }
    if (tid == 0) partials[blockIdx.x] = red[0];
}

// ---------------------------------------------------------------------------
// Final reduction of NPART per-block partials with V_WMMA_F32_16X16X4_F32:
// A (16x4 f32, 2 VGPRs/lane) holds 64 lane accumulators, B = ones(4x16),
// C = 0  =>  D[m,n] = rowsum(A)[m].  Per the 16x16 f32 C/D layout, lane 0
// holds D[0..7,0] and lane 16 holds D[8..15,0]; one shfl_xor(16) finishes
// the exact FP32 sum. Wave is launched with 32 threads -> EXEC all ones.
// ---------------------------------------------------------------------------
typedef __attribute__((ext_vector_type(2))) float v2f;
typedef __attribute__((ext_vector_type(8))) float v8f;

__global__ __launch_bounds__(32) void yolo_final(
    const float* __restrict__ partials, int n, float* __restrict__ out) {
    const int lane = threadIdx.x;
    float a0 = 0.0f, a1 = 0.0f;
    for (int i = lane; i < n; i += 64) a0 += partials[i];
    for (int i = lane + 32; i < n; i += 64) a1 += partials[i];

    v2f A;
    A[0] = a0;
    A[1] = a1;
    v2f B;
    B[0] = 1.0f;
    B[1] = 1.0f;
    v8f C = {};
    v8f D = __builtin_amdgcn_wmma_f32_16x16x4_f32(
        /*neg_a=*/false, A, /*neg_b=*/false, B,
        /*c_mod=*/(short)0, C, /*reuse_a=*/false, /*reuse_b=*/false);

    float s = D[0] + D[1] + D[2] + D[3] + D[4] + D[5] + D[6] + D[7];
    float tot = s + __shfl_xor(s, 16, 32);
    if (lane == 0) out[0] = tot;
}

extern "C" void kernel_launch(void* const* d_in, const int* in_sizes, int n_in,
                              void* d_out, int out_size, void* d_ws,
                              size_t ws_size, hipStream_t stream) {
    // setup_inputs() dict order:
    // 0:p0 1:ri0 2:rj0 3:rk0 4:p1 5:ri1 6:rj1 7:rk1 8:p2 9:ri2 10:rj2 11:rk2
    // 12:boxes 13:labels 14:strides
    const float* p0    = (const float*)d_in[0];
    const int* ri0     = (const int*)d_in[1];
    const int* rj0     = (const int*)d_in[2];
    const int* rk0     = (const int*)d_in[3];
    const float* p1    = (const float*)d_in[4];
    const int* ri1     = (const int*)d_in[5];
    const int* rj1     = (const int*)d_in[6];
    const int* rk1     = (const int*)d_in[7];
    const float* p2    = (const float*)d_in[8];
    const int* ri2     = (const int*)d_in[9];
    const int* rj2     = (const int*)d_in[10];
    const int* rk2     = (const int*)d_in[11];
    const float* boxes = (const float*)d_in[12];
    const int* labels  = (const int*)d_in[13];
    const int* strides = (const int*)d_in[14];

    float* ws  = (float*)d_ws;
    float* out = (float*)d_out;

    // partials: [0,64) L0 pos | [64,128) L1 pos | [128,192) L2 pos |
    //           [192,380) negatives — every slot rewritten each call.
    yolo_pos<<<POS_BLOCKS_PER_LVL, 256, 0, stream>>>(p0, boxes, labels, strides,
                                                     0, 80, 80, ws + 0);
    yolo_pos<<<POS_BLOCKS_PER_LVL, 256, 0, stream>>>(p1, boxes, labels, strides,
                                                     1, 40, 40, ws + 64);
    yolo_pos<<<POS_BLOCKS_PER_LVL, 256, 0, stream>>>(p2, boxes, labels, strides,
                                                     2, 20, 20, ws + 128);
    yolo_neg<<<NEG_BLOCKS, 256, 0, stream>>>(p0, p1, p2, ri0, rj0, rk0, ri1,
                                             rj1, rk1, ri2, rj2, rk2, ws + 192);
    yolo_final<<<1, 32, 0, stream>>>(ws, NPART, out);
}